// GraphConv_45483703664784
// MI455X (gfx1250) — compile-verified
//
#include <hip/hip_runtime.h>

// GraphConv (GCN) for MI455X / gfx1250, wave32.
//   norm = rsqrt(clamp(out_degree,1))
//   h    = (feat @ W) * norm[:,None]        (V_WMMA_F32_16X16X4_F32)
//   out  = scatter_add(h[src] -> dst) * norm[:,None] + bias

#define N_NODES 50000
#define N_EDGES 800000
#define IN_F    256
#define OUT_F   64

#define LDS_PITCH 260   // 256 + 4 floats: de-conflicts LDS banks for per-row K reads

typedef float v2f __attribute__((ext_vector_type(2)));
typedef float v8f __attribute__((ext_vector_type(8)));

// ---------------------------------------------------------------- utilities
__global__ void zero_f32(float* __restrict__ p, int n) {
    int i = blockIdx.x * blockDim.x + threadIdx.x;
    if (i < n) p[i] = 0.0f;
}

// deg[src[e]] += 1  (out-degree per segment_sum over src)
__global__ void degree_kernel(const int* __restrict__ ei, float* __restrict__ deg) {
    int e = blockIdx.x * blockDim.x + threadIdx.x;
    if (e < N_EDGES) atomicAdd(&deg[ei[e]], 1.0f);
}

// in-place: deg -> rsqrt(max(deg,1))
__global__ void norm_kernel(float* __restrict__ dn) {
    int i = blockIdx.x * blockDim.x + threadIdx.x;
    if (i < N_NODES) {
        float d = dn[i];
        d = d < 1.0f ? 1.0f : d;
        dn[i] = rsqrtf(d);
    }
}

// ---------------------------------------------------------------- WMMA GEMM
// Block: 128 threads = 4 wave32; block computes rows [m0, m0+16) x all 64 cols.
// Wave w computes the 16-col slice n0 = 16*w, accumulating K=256 in steps of 4
// via V_WMMA_F32_16X16X4_F32 with 4 interleaved accumulators.
__global__ __launch_bounds__(128)
void gemm_wmma_f32(const float* __restrict__ feat,
                   const float* __restrict__ W,
                   const float* __restrict__ norm,
                   float* __restrict__ h) {
    __shared__ float As[16 * LDS_PITCH];

    const int tid  = threadIdx.x;
    const int wave = tid >> 5;
    const int lane = tid & 31;
    const int m0   = blockIdx.x * 16;
    const int n0   = wave * 16;

    // Cooperative, coalesced stage of the 16x256 feat tile into LDS.
    for (int idx = tid; idx < 16 * IN_F; idx += 128) {
        const int r = idx >> 8;      // row 0..15
        const int c = idx & 255;     // col 0..255
        As[r * LDS_PITCH + c] = feat[(m0 + r) * IN_F + c];
    }
    __syncthreads();

    const int lg = lane >> 4;   // lane group: 0 -> K+{0,1}, 1 -> K+{2,3}
    const int ll = lane & 15;   // A: row index / B,C,D: col index

    v8f acc[4];
    acc[0] = v8f{}; acc[1] = v8f{}; acc[2] = v8f{}; acc[3] = v8f{};

    for (int k0 = 0; k0 < IN_F; k0 += 16) {
#pragma unroll
        for (int u = 0; u < 4; ++u) {
            const int k = k0 + 4 * u;
            // A-frag (16x4 f32): lane ll holds row ll, K = k+2*lg .. +1 (aligned v2f)
            v2f a = *(const v2f*)(&As[ll * LDS_PITCH + k + 2 * lg]);
            // B-frag (4x16 f32): lane ll holds col n0+ll, VGPR j -> K = k+2*lg+j
            v2f b;
            b.x = W[(k + 2 * lg + 0) * OUT_F + n0 + ll];
            b.y = W[(k + 2 * lg + 1) * OUT_F + n0 + ll];
            acc[u] = __builtin_amdgcn_wmma_f32_16x16x4_f32(
                /*neg_a=*/false, a, /*neg_b=*/false, b,
                /*c_mod=*/(short)0, acc[u],
                /*reuse_a=*/false, /*reuse_b=*/false);
        }
    }

    v8f c = acc[0] + acc[1] + acc[2] + acc[3];

    // D layout: element r of lane L -> row m0 + r + 8*lg, col n0 + ll.
    // Fold the source-side norm scaling into the store (row-constant scale
    // commutes with the GEMM).
#pragma unroll
    for (int r = 0; r < 8; ++r) {
        const int row = m0 + r + 8 * lg;
        h[row * OUT_F + n0 + ll] = c[r] * norm[row];
    }
}

// ---------------------------------------------------------------- scatter
// One thread per (edge, feature). Each 64-thread group reads a contiguous
// 256B row of h and issues 64 contiguous global_atomic_add_f32 into out.
__global__ void scatter_kernel(const int* __restrict__ ei,
                               const float* __restrict__ h,
                               float* __restrict__ out) {
    int idx = blockIdx.x * blockDim.x + threadIdx.x;   // < 51.2M, fits int32
    if (idx >= N_EDGES * OUT_F) return;
    const int e = idx >> 6;
    const int f = idx & 63;
    const int src = ei[e];
    const int dst = ei[N_EDGES + e];
    atomicAdd(&out[dst * OUT_F + f], h[src * OUT_F + f]);
}

// out = out * norm[node] + bias[f]
__global__ void finalize_kernel(const float* __restrict__ norm,
                                const float* __restrict__ bias,
                                float* __restrict__ out) {
    int i = blockIdx.x * blockDim.x + threadIdx.x;
    if (i < N_NODES * OUT_F) {
        const int node = i >> 6;
        const int f    = i & 63;
        out[i] = out[i] * norm[node] + bias[f];
    }
}

// ---------------------------------------------------------------- launcher
extern "C" void kernel_launch(void* const* d_in, const int* in_sizes, int n_in,
                              void* d_out, int out_size, void* d_ws, size_t ws_size,
                              hipStream_t stream) {
    const float* feat = (const float*)d_in[0];   // [50000, 256]
    const float* W    = (const float*)d_in[1];   // [256, 64]
    const float* bias = (const float*)d_in[2];   // [64]
    const int*   ei   = (const int*)d_in[3];     // [2, 800000]
    float* out = (float*)d_out;                  // [50000, 64]

    // Workspace layout: norm (50048 f32, padded for 256B alignment) | h (50000*64 f32)
    float* norm = (float*)d_ws;
    float* h    = norm + 50048;

    // 1) zero degree buffer and output accumulator (deterministic per call)
    zero_f32<<<(N_NODES + 255) / 256, 256, 0, stream>>>(norm, N_NODES);
    zero_f32<<<(N_NODES * OUT_F + 255) / 256, 256, 0, stream>>>(out, N_NODES * OUT_F);

    // 2) out-degree via f32 atomics
    degree_kernel<<<(N_EDGES + 255) / 256, 256, 0, stream>>>(ei, norm);

    // 3) norm = rsqrt(clamp(deg, 1))
    norm_kernel<<<(N_NODES + 255) / 256, 256, 0, stream>>>(norm);

    // 4) h = (feat @ W) * norm   -- WMMA f32 16x16x4, 3125 blocks x 4 waves
    gemm_wmma_f32<<<N_NODES / 16, 128, 0, stream>>>(feat, W, norm, h);

    // 5) out[dst] += h[src]  (51.2M coalesced f32 atomics, L2-resident)
    scatter_kernel<<<(N_EDGES * OUT_F) / 256, 256, 0, stream>>>(ei, h, out);

    // 6) out = out * norm + bias
    finalize_kernel<<<(N_NODES * OUT_F + 255) / 256, 256, 0, stream>>>(norm, bias, out);
}